// GraphAttentionLayer_30812095381683
// MI455X (gfx1250) — compile-verified
//
#include <hip/hip_runtime.h>

// ---------------------------------------------------------------------------
// GAT layer, fused flash-style for MI455X (gfx1250, wave32, WMMA).
//   B=4, N=4096, F_in=128, F_out=64
// Memory floor: adj = 256MB read once @23.3TB/s ~ 11us. Wh (1MB/batch) stays
// in the 192MB L2, so the 1GB of logical Wh re-reads are L2 hits.
// Matmul work (8.6 GFLOP) runs on V_WMMA_F32_16X16X4_F32 to keep full fp32
// accuracy on the 4096-term weighted sums.
// ---------------------------------------------------------------------------

typedef float v2f __attribute__((ext_vector_type(2)));
typedef float v8f __attribute__((ext_vector_type(8)));

#define LEAKY 0.2f
#define EPSV 1e-9f

constexpr int Bn = 4, Nn = 4096, Fin = 128, Fout = 64;
constexpr int ROWS = Bn * Nn; // 16384 flattened rows

__device__ __forceinline__ float lrelu(float x) { return fmaxf(x, LEAKY * x); }

// ---------------------------------------------------------------------------
__global__ void k0_init(unsigned int* menc) {
  if (threadIdx.x < Bn) menc[threadIdx.x] = 0u; // 0 <= every monotone encoding
}

// ---------------------------------------------------------------------------
// Wh = h @ W. One block = one 16-row tile; wave w owns 16-col tile w (of 4).
// K loop: 32 steps of V_WMMA_F32_16X16X4_F32.
__global__ void k1_gemm(const float* __restrict__ h, const float* __restrict__ W,
                        float* __restrict__ wh) {
  const int lane = threadIdx.x & 31;
  const int wv   = threadIdx.x >> 5;
  const int l15  = lane & 15;
  const int hi   = lane >> 4; // half-wave: K pair select (0->K0,1 ; 1->K2,3)
  const int row0 = blockIdx.x * 16;
  const int col0 = wv * 16;

  v8f acc = {};
  const float* hrow = h + (size_t)(row0 + l15) * Fin + 2 * hi;
  for (int k = 0; k < Fin; k += 4) {
    // A 16x4 f32: lane l15 = row; VGPR0/1 = K 2*hi / 2*hi+1
    v2f a = *(const v2f*)(hrow + k);
    // B 4x16 f32: lane l15 = col; VGPR0/1 = K 2*hi / 2*hi+1
    const float* wp = W + (size_t)(k + 2 * hi) * Fout + col0 + l15;
    v2f b;
    b.x = wp[0];
    b.y = wp[Fout];
    acc = __builtin_amdgcn_wmma_f32_16x16x4_f32(false, a, false, b, (short)0,
                                                acc, false, false);
  }
  // C layout: VGPR r holds row r + 8*hi, col = l15
  float* outp = wh + (size_t)(row0 + 8 * hi) * Fout + col0 + l15;
#pragma unroll
  for (int r = 0; r < 8; ++r) outp[(size_t)r * Fout] = acc[r];
}

// ---------------------------------------------------------------------------
// s1 = Wh@a1, s2 = Wh@a2, plus per-batch max_j s1 via monotone-uint atomicMax.
// One wave per row (64-length dot, 2 elems/lane, xor-shuffle reduce).
__global__ void k2_scores(const float* __restrict__ wh, const float* __restrict__ a,
                          float* __restrict__ s1, float* __restrict__ s2,
                          unsigned int* __restrict__ menc) {
  const int lane = threadIdx.x & 31;
  const int row  = blockIdx.x * 8 + (threadIdx.x >> 5);
  const float w0 = wh[(size_t)row * Fout + lane];
  const float w1 = wh[(size_t)row * Fout + 32 + lane];
  float d1 = w0 * a[lane] + w1 * a[32 + lane];        // a1 = a[0:64]
  float d2 = w0 * a[64 + lane] + w1 * a[96 + lane];   // a2 = a[64:128]
#pragma unroll
  for (int m = 16; m > 0; m >>= 1) {
    d1 += __shfl_xor(d1, m, 32);
    d2 += __shfl_xor(d2, m, 32);
  }
  if (lane == 0) {
    s1[row] = d1;
    s2[row] = d2;
    unsigned int u = __float_as_uint(d1);
    u = (u & 0x80000000u) ? ~u : (u | 0x80000000u); // order-preserving encode
    atomicMax(&menc[row >> 12], u);                 // row / 4096 -> batch
  }
}

// ---------------------------------------------------------------------------
// Fused: stream adj once; per 16-row tile accumulate
//   C[i,:] += q_ij * Wh[j,:],  Z_i += p_ij,  S_i += q_ij
// with p = exp(lrelu(s2_i+s1_j) - m_i), q = p*adj. Final: C / (S + eps*Z).
// Block = 4 waves on the SAME 16 rows; wave w takes K-slices w, w+4, ...
// so we get 4096 concurrent waves; partials combined through LDS.
__global__ void k3_attn(const float* __restrict__ adj, const float* __restrict__ wh,
                        const float* __restrict__ s1, const float* __restrict__ s2,
                        const unsigned int* __restrict__ menc,
                        float* __restrict__ out) {
  __shared__ float redC[4][16 * 64];
  __shared__ float redZ[4][16];
  __shared__ float redS[4][16];

  const int lane = threadIdx.x & 31;
  const int wv   = threadIdx.x >> 5;
  const int l15  = lane & 15;
  const int hi   = lane >> 4;
  const int tile = blockIdx.x;        // 0..1023
  const int b    = tile >> 8;         // 256 tiles per batch
  const int row0 = tile * 16;         // flattened row base
  const int rowg = row0 + l15;

  // exact row max: m_i = lrelu(s2_i + max_j s1_j)  (lrelu is monotone)
  const unsigned int me = menc[b];
  const float Mb = (me & 0x80000000u) ? __uint_as_float(me ^ 0x80000000u)
                                      : __uint_as_float(~me);
  const float s2r = s2[rowg];
  const float mr  = lrelu(s2r + Mb);

  const float* adjRow = adj + ((size_t)b << 24) + (size_t)(rowg & (Nn - 1)) * Nn;
  const float* s1b    = s1 + (size_t)b * Nn;
  const float* whB    = wh + (size_t)b * Nn * Fout;

  v8f acc0 = {}, acc1 = {}, acc2 = {}, acc3 = {};
  float zacc = 0.f, sacc = 0.f;

  for (int ks = wv; ks < Nn / 4; ks += 4) {
    const int j = ks * 4 + 2 * hi; // this lane's K pair
    // ---- A fragment: attention weights computed on the fly
    v2f ad = *(const v2f*)(adjRow + j); // adj[b, row, j], adj[b, row, j+1]
    v2f sv = *(const v2f*)(s1b + j);
    float p0 = __expf(lrelu(s2r + sv.x) - mr);
    float p1 = __expf(lrelu(s2r + sv.y) - mr);
    zacc += p0 + p1;
    float q0 = p0 * ad.x, q1 = p1 * ad.y;
    sacc += q0 + q1;
    v2f afrag;
    afrag.x = q0;
    afrag.y = q1;
    // ---- B fragments: Wh rows j, j+1 (L2-resident), 4 col tiles
    const float* wrow = whB + (size_t)j * Fout + l15;
    v2f b0, b1, b2, b3;
    b0.x = wrow[0];  b0.y = wrow[Fout];
    b1.x = wrow[16]; b1.y = wrow[16 + Fout];
    b2.x = wrow[32]; b2.y = wrow[32 + Fout];
    b3.x = wrow[48]; b3.y = wrow[48 + Fout];
    acc0 = __builtin_amdgcn_wmma_f32_16x16x4_f32(false, afrag, false, b0, (short)0, acc0, false, false);
    acc1 = __builtin_amdgcn_wmma_f32_16x16x4_f32(false, afrag, false, b1, (short)0, acc1, false, false);
    acc2 = __builtin_amdgcn_wmma_f32_16x16x4_f32(false, afrag, false, b2, (short)0, acc2, false, false);
    acc3 = __builtin_amdgcn_wmma_f32_16x16x4_f32(false, afrag, false, b3, (short)0, acc3, false, false);
  }

  // lanes l and l+16 covered complementary j's of the same row
  zacc += __shfl_xor(zacc, 16, 32);
  sacc += __shfl_xor(sacc, 16, 32);
  if (hi == 0) {
    redZ[wv][l15] = zacc;
    redS[wv][l15] = sacc;
  }
  {
    float* p = &redC[wv][0];
#pragma unroll
    for (int r = 0; r < 8; ++r) {
      const int row = r + 8 * hi; // C layout: VGPR r -> row r + 8*hi
      p[row * 64 + 0  + l15] = acc0[r];
      p[row * 64 + 16 + l15] = acc1[r];
      p[row * 64 + 32 + l15] = acc2[r];
      p[row * 64 + 48 + l15] = acc3[r];
    }
  }
  __syncthreads();

  // combine the 4 wave-partials, normalize, coalesced store
  float* outB = out + (size_t)row0 * Fout;
  for (int e = threadIdx.x; e < 16 * 64; e += 128) {
    const int row = e >> 6;
    float v = redC[0][e] + redC[1][e] + redC[2][e] + redC[3][e];
    float z = redZ[0][row] + redZ[1][row] + redZ[2][row] + redZ[3][row];
    float s = redS[0][row] + redS[1][row] + redS[2][row] + redS[3][row];
    outB[e] = v / (s + EPSV * z);
  }
}

// ---------------------------------------------------------------------------
extern "C" void kernel_launch(void* const* d_in, const int* in_sizes, int n_in,
                              void* d_out, int out_size, void* d_ws, size_t ws_size,
                              hipStream_t stream) {
  const float* h   = (const float*)d_in[0]; // (4,4096,128)
  const float* adj = (const float*)d_in[1]; // (4,4096,4096)
  const float* W   = (const float*)d_in[2]; // (128,64)
  const float* a   = (const float*)d_in[3]; // (128,1)
  float* out = (float*)d_out;               // (4,4096,64)

  // workspace layout (floats): Wh[16384*64] | s1[16384] | s2[16384] | menc[4]
  float* wh = (float*)d_ws;
  float* s1 = wh + (size_t)ROWS * Fout;
  float* s2 = s1 + ROWS;
  unsigned int* menc = (unsigned int*)(s2 + ROWS);

  k0_init<<<1, 32, 0, stream>>>(menc);
  k1_gemm<<<ROWS / 16, 128, 0, stream>>>(h, W, wh);
  k2_scores<<<ROWS / 8, 256, 0, stream>>>(wh, a, s1, s2, menc);
  k3_attn<<<ROWS / 16, 128, 0, stream>>>(adj, wh, s1, s2, menc, out);
}